// GCN_Concat_5342939316780
// MI455X (gfx1250) — compile-verified
//
#include <hip/hip_runtime.h>

typedef __attribute__((ext_vector_type(2))) float v2f;
typedef __attribute__((ext_vector_type(8))) float v8f;

#define HIDW 128
#define KCHUNK 32

// ---------------- degree / norm ----------------
__global__ __launch_bounds__(256) void k_fill1(float* p, int n) {
  int i = blockIdx.x * blockDim.x + threadIdx.x;
  if (i < n) p[i] = 1.0f;   // self-loop contribution to degree
}

__global__ __launch_bounds__(256) void k_edge_deg(float* deg, const int* __restrict__ dst, int E) {
  int e = blockIdx.x * blockDim.x + threadIdx.x;
  if (e < E) atomicAdd(&deg[dst[e]], 1.0f);
}

__global__ __launch_bounds__(256) void k_dis(float* dis, const float* __restrict__ deg, int n) {
  int i = blockIdx.x * blockDim.x + threadIdx.x;
  if (i < n) {
    float d = deg[i];
    dis[i] = (d > 0.f) ? rsqrtf(d) : 0.f;
  }
}

// ---------------- propagation ----------------
// out[i,f] = h[i,f]*dis[i]^2 + bias[f]   (self-loop term + folded bias)
__global__ __launch_bounds__(256) void k_prop_init(float* __restrict__ out, const float* __restrict__ h,
                                                   const float* __restrict__ dis, const float* __restrict__ bias,
                                                   int n) {
  int idx = blockIdx.x * blockDim.x + threadIdx.x;
  if (idx < n * HIDW) {
    int i = idx >> 7;
    int f = idx & 127;
    float di = dis[i];
    out[idx] = h[idx] * di * di + bias[f];
  }
}

// out[dst,f] += h[src,f] * dis[src]*dis[dst]; 2 edges per 256-thread block
__global__ __launch_bounds__(256) void k_prop_scatter(const float* __restrict__ h, const int* __restrict__ src,
                                                      const int* __restrict__ dst, const float* __restrict__ dis,
                                                      float* __restrict__ out, int E) {
  int e = blockIdx.x * 2 + (threadIdx.x >> 7);
  if (e >= E) return;
  int f = threadIdx.x & 127;
  int s = src[e], d = dst[e];
  float nrm = dis[s] * dis[d];
  atomicAdd(&out[(size_t)d * HIDW + f], h[(size_t)s * HIDW + f] * nrm);
}

// ---------------- WMMA fp32 GEMM ----------------
// C[M,NC] = act( [A0 | A1][M, KW*128] @ W[NC, KW*128]^T + bias )
// block: 256 threads = 8 waves; tile 128 rows x NC cols; wave w owns rows w*16..w*16+15.
template<int NC, int KW, bool RELU_A0, bool RELU_OUT, bool HAS_BIAS>
__global__ __launch_bounds__(256) void k_gemm(const float* __restrict__ A0, const float* __restrict__ A1,
                                              const float* __restrict__ W, const float* __restrict__ bias,
                                              float* __restrict__ C, int M) {
  __shared__ __align__(16) float As[128][KCHUNK + 4];  // stride 36: 16B aligned, conflict-free cols
  __shared__ __align__(16) float Ws[NC][KCHUNK + 4];
  constexpr int NT = NC / 16;
  constexpr int K = KW * 128;

  const int tid  = threadIdx.x;
  const int wave = tid >> 5;
  const int lane = tid & 31;
  const int l15  = lane & 15;
  const int half = lane >> 4;
  const int m0   = blockIdx.x * 128;

  v8f acc[NT];
#pragma unroll
  for (int t = 0; t < NT; ++t)
#pragma unroll
    for (int r = 0; r < 8; ++r) acc[t][r] = 0.0f;

  for (int kb = 0; kb < K; kb += KCHUNK) {
    const float* Ablk = (KW == 2 && kb >= 128) ? A1 : A0;
    const bool doRelu = RELU_A0 && (kb < 128);
    const int kloc = kb & 127;

    // stage A chunk: 128 rows x 32 cols, float4 per thread per round
#pragma unroll
    for (int j = 0; j < (128 * KCHUNK) / 1024; ++j) {
      int ei  = tid * 4 + j * 1024;
      int row = ei >> 5;
      int col = ei & 31;
      int grow = m0 + row;
      float4 v = {0.f, 0.f, 0.f, 0.f};
      if (grow < M) v = *(const float4*)&Ablk[(size_t)grow * HIDW + kloc + col];
      if (doRelu) {
        v.x = fmaxf(v.x, 0.f); v.y = fmaxf(v.y, 0.f);
        v.z = fmaxf(v.z, 0.f); v.w = fmaxf(v.w, 0.f);
      }
      *(float4*)&As[row][col] = v;
    }
    // stage W chunk: NC rows x 32 cols
#pragma unroll
    for (int j = 0; j < (NC * KCHUNK) / 1024; ++j) {
      int ei  = tid * 4 + j * 1024;
      int n   = ei >> 5;
      int col = ei & 31;
      *(float4*)&Ws[n][col] = *(const float4*)&W[(size_t)n * K + kb + col];
    }
    __syncthreads();

    const int arow = wave * 16 + l15;
    const int o = half * 2;  // lanes 0-15: K=k,k+1 ; lanes 16-31: K=k+2,k+3
#pragma unroll
    for (int k4 = 0; k4 < KCHUNK; k4 += 4) {
      v2f a = *(const v2f*)&As[arow][k4 + o];
#pragma unroll
      for (int t = 0; t < NT; ++t) {
        v2f b = *(const v2f*)&Ws[t * 16 + l15][k4 + o];
        acc[t] = __builtin_amdgcn_wmma_f32_16x16x4_f32(
            /*neg_a=*/false, a, /*neg_b=*/false, b,
            /*c_mod=*/(short)0, acc[t], /*reuse_a=*/false, /*reuse_b=*/false);
      }
    }
    __syncthreads();
  }

  // epilogue: lane l, vgpr r -> row = m0 + wave*16 + half*8 + r, col = t*16 + (lane&15)
  const int rbase = m0 + wave * 16 + half * 8;
#pragma unroll
  for (int t = 0; t < NT; ++t) {
    int col = t * 16 + l15;
    float bv = HAS_BIAS ? bias[col] : 0.f;
#pragma unroll
    for (int r = 0; r < 8; ++r) {
      int row = rbase + r;
      if (row < M) {
        float v = acc[t][r] + bv;
        if (RELU_OUT) v = fmaxf(v, 0.f);
        C[(size_t)row * NC + col] = v;
      }
    }
  }
}

// ---------------- launch ----------------
extern "C" void kernel_launch(void* const* d_in, const int* in_sizes, int n_in,
                              void* d_out, int out_size, void* d_ws, size_t ws_size,
                              hipStream_t stream) {
  const float* x  = (const float*)d_in[0];
  const int* edge = (const int*)d_in[1];   // [2, E] int32: row0 = src, row1 = dst
  const float* W1 = (const float*)d_in[2];
  const float* b1 = (const float*)d_in[3];
  const float* Wc = (const float*)d_in[4];
  const float* bc = (const float*)d_in[5];
  const float* W2 = (const float*)d_in[6];
  const float* b2 = (const float*)d_in[7];
  const float* Wo = (const float*)d_in[8];
  const float* bo = (const float*)d_in[9];

  const int N = in_sizes[0] / HIDW;
  const int E = in_sizes[1] / 2;
  const int* srcIdx = edge;
  const int* dstIdx = edge + E;

  float* wsf = (float*)d_ws;
  size_t NP   = (((size_t)N) + 255) & ~(size_t)255;
  float* dis  = wsf;                       // [N]
  float* buf1 = wsf + NP;                  // h / h2          [N,128]
  float* buf2 = buf1 + (size_t)N * HIDW;   // x_agg / x_agg2  [N,128]
  float* buf3 = buf2 + (size_t)N * HIDW;   // x_comb          [N,128]
  float* deg  = buf1;                      // reuse buf1 before first GEMM

  dim3 b256(256);
  int gN  = (N + 255) / 256;
  int gNH = (N * HIDW + 255) / 256;
  int gE  = (E + 255) / 256;
  int gE2 = (E + 1) / 2;
  int gG  = (N + 127) / 128;

  // norm coefficients
  k_fill1   <<<gN,  b256, 0, stream>>>(deg, N);
  k_edge_deg<<<gE,  b256, 0, stream>>>(deg, dstIdx, E);
  k_dis     <<<gN,  b256, 0, stream>>>(dis, deg, N);

  // layer 1: h = x @ W1^T ; x_agg = scatter(h, norm) + b1  (relu deferred to next GEMM's A-load)
  k_gemm<128,1,false,false,false><<<gG, b256, 0, stream>>>(x, nullptr, W1, nullptr, buf1, N);
  k_prop_init   <<<gNH, b256, 0, stream>>>(buf2, buf1, dis, b1, N);
  k_prop_scatter<<<gE2, b256, 0, stream>>>(buf1, srcIdx, dstIdx, dis, buf2, E);

  // combine: x_comb = relu([relu(x_agg) | x] @ Wc^T + bc)
  k_gemm<128,2,true,true,true><<<gG, b256, 0, stream>>>(buf2, x, Wc, bc, buf3, N);

  // layer 2: h2 = x_comb @ W2^T ; x_agg2 = scatter(h2, norm) + b2
  k_gemm<128,1,false,false,false><<<gG, b256, 0, stream>>>(buf3, nullptr, W2, nullptr, buf1, N);
  k_prop_init   <<<gNH, b256, 0, stream>>>(buf2, buf1, dis, b2, N);
  k_prop_scatter<<<gE2, b256, 0, stream>>>(buf1, srcIdx, dstIdx, dis, buf2, E);

  // output: out = [x_agg2 | x_comb] @ Wo^T + bo
  k_gemm<64,2,false,false,true><<<gG, b256, 0, stream>>>(buf2, buf3, Wo, bo, (float*)d_out, N);
}